// Conv2d_53334903882373
// MI455X (gfx1250) — compile-verified
//
#include <hip/hip_runtime.h>

// ---------------------------------------------------------------------------
// Fused Winograd F(2x2,3x3) convolution for MI455X (gfx1250, wave32).
//   x: (8,64,128,128) f32, filt: (64,64,3,3) f32 -> Y: (8,64,126,126) f32
//
// Kernel A: filter transform U = G g G^T, stored pre-permuted into the exact
//           v_wmma_f32_16x16x4_f32 A-fragment layout (coalesced b64 loads).
// Kernel B: fully fused per-block pipeline in LDS (304 KB of the 320 KB/WGP):
//           input transform -> Vs, 16 GEMMs via WMMA -> Ms, inverse transform
//           -> Y.  V and M never touch global memory.
// ---------------------------------------------------------------------------

#define WN   8
#define WC   64
#define WK   64
#define WH   128
#define WW   128
#define WT   63                       // tiles per spatial dim
#define WP   (WN * WT * WT)           // 31752 GEMM columns
#define WOH  126
#define TP   32                       // GEMM columns per block
#define NBLK ((WP + TP - 1) / TP)     // 993

#define VS_STRIDE 40                  // 2*40 % 64 == 16 -> conflict-free B reads
#define MS_STRIDE 36                  // 8*36 % 64 == 32 -> conflict-free D stores
#define VS_FLOATS (16 * WC * VS_STRIDE)           // 40960
#define MS_FLOATS (16 * WK * MS_STRIDE)           // 36864
#define SMEM_BYTES ((VS_FLOATS + MS_FLOATS) * 4)  // 311296 <= 327680

typedef float v2f __attribute__((ext_vector_type(2)));
typedef float v8f __attribute__((ext_vector_type(8)));

// ---------------------------------------------------------------------------
// Filter transform. Output A2 is the A-operand fragment table:
//   A2[(((ab*4 + krow)*16 + cstep)*64 + lane)*2 + e]
//     = U[ab][krow*16 + (lane%16)][cstep*4 + (lane/16)*2 + e]
// so in the GEMM each lane issues ONE b64 load per wmma, fully coalesced.
// ---------------------------------------------------------------------------
__global__ __launch_bounds__(256) void wg_filter(const float* __restrict__ g,
                                                 float* __restrict__ A2) {
    int tid = blockIdx.x * blockDim.x + threadIdx.x;
    if (tid >= WK * WC) return;
    int k = tid / WC, c = tid % WC;
    const float* gp = g + (size_t)(k * WC + c) * 9;
    float d[3][3];
#pragma unroll
    for (int i = 0; i < 3; ++i)
#pragma unroll
        for (int j = 0; j < 3; ++j) d[i][j] = gp[i * 3 + j];

    float t[4][3];
#pragma unroll
    for (int j = 0; j < 3; ++j) {
        t[0][j] = d[0][j];
        t[1][j] = 0.5f * (d[0][j] + d[1][j] + d[2][j]);
        t[2][j] = 0.5f * (d[0][j] - d[1][j] + d[2][j]);
        t[3][j] = d[2][j];
    }
    float u[4][4];
#pragma unroll
    for (int a = 0; a < 4; ++a) {
        u[a][0] = t[a][0];
        u[a][1] = 0.5f * (t[a][0] + t[a][1] + t[a][2]);
        u[a][2] = 0.5f * (t[a][0] - t[a][1] + t[a][2]);
        u[a][3] = t[a][2];
    }

    int krow = k >> 4, lnn = k & 15;
    int cstep = c >> 2, rem = c & 3, hf = rem >> 1, e = rem & 1;
    int lanei = hf * 16 + lnn;
#pragma unroll
    for (int a = 0; a < 4; ++a)
#pragma unroll
        for (int b = 0; b < 4; ++b)
            A2[(size_t)((((a * 4 + b) * 4 + krow) * 16 + cstep) * 64 + lanei) * 2 + e]
                = u[a][b];
}

// ---------------------------------------------------------------------------
// Fused main kernel: one block handles TP=32 GEMM columns end-to-end.
// ---------------------------------------------------------------------------
__global__ __launch_bounds__(256) void wg_fused(const float* __restrict__ x,
                                                const float* __restrict__ A2,
                                                float* __restrict__ Y) {
    extern __shared__ float smem[];
    float* Vs = smem;                 // [ab][c][VS_STRIDE]
    float* Ms = smem + VS_FLOATS;     // [ab][k][MS_STRIDE]

    const int tid = threadIdx.x;
    const int p0  = blockIdx.x * TP;

    // ---- stage 1: input transform V = BT d BT^T  -> Vs ---------------------
    // 2048 (c,pl) items, 8 per thread; 32 consecutive threads share c and
    // sweep pl 0..31 (coalesced-ish x reads, conflict-free Vs writes).
#pragma unroll
    for (int it = 0; it < 8; ++it) {
        int item = it * 256 + tid;
        int pl = item & 31;
        int c  = item >> 5;
        int p  = p0 + pl;
        float v[4][4];
        if (p < WP) {
            int n = p / (WT * WT);
            int r = p % (WT * WT);
            int t = r / WT, uu = r % WT;
            const float* xp = x + (((size_t)(n * WC + c) * WH) + 2 * t) * WW + 2 * uu;
            float d[4][4];
#pragma unroll
            for (int i = 0; i < 4; ++i)
#pragma unroll
                for (int j = 0; j < 4; ++j) d[i][j] = xp[i * WW + j];
            float w[4][4];
#pragma unroll
            for (int j = 0; j < 4; ++j) {
                w[0][j] = d[0][j] - d[2][j];
                w[1][j] = d[1][j] + d[2][j];
                w[2][j] = d[2][j] - d[1][j];
                w[3][j] = d[1][j] - d[3][j];
            }
#pragma unroll
            for (int a = 0; a < 4; ++a) {
                v[a][0] = w[a][0] - w[a][2];
                v[a][1] = w[a][1] + w[a][2];
                v[a][2] = w[a][2] - w[a][1];
                v[a][3] = w[a][1] - w[a][3];
            }
        } else {
#pragma unroll
            for (int a = 0; a < 4; ++a)
#pragma unroll
                for (int b = 0; b < 4; ++b) v[a][b] = 0.0f;
        }
#pragma unroll
        for (int a = 0; a < 4; ++a)
#pragma unroll
            for (int b = 0; b < 4; ++b)
                Vs[((a * 4 + b) * WC + c) * VS_STRIDE + pl] = v[a][b];
    }
    __syncthreads();

    // ---- stage 2: 16 GEMMs  Ms[ab] = U[ab](64x64) x Vs[ab](64x32) ----------
    // 128 wave-tile-tasks (16 ab x 4 krow x 2 ptile), 16 per wave,
    // 16 v_wmma_f32_16x16x4_f32 each.  Wave-uniform control flow (EXEC all-1s).
    const int wave = tid >> 5;
    const int lane = tid & 31;
    const int half = lane >> 4;
    const int ln   = lane & 15;
    const int koff = half * 2;

    for (int task = wave; task < 128; task += 8) {
        int ab   = task >> 3;
        int krow = (task >> 1) & 3;
        int pt   = (task & 1) * 16;
        const float* ap = A2 + (size_t)(((ab * 4 + krow) * 16) * 64 + lane * 2);
        const float* vb = Vs + (ab * WC) * VS_STRIDE + pt + ln;
        v8f acc = {};
#pragma unroll
        for (int cs = 0; cs < 16; ++cs) {
            v2f a = *(const v2f*)(ap + cs * 128);   // one coalesced b64 / lane
            v2f b;
            b.x = vb[(cs * 4 + koff)     * VS_STRIDE];
            b.y = vb[(cs * 4 + koff + 1) * VS_STRIDE];
            acc = __builtin_amdgcn_wmma_f32_16x16x4_f32(false, a, false, b,
                                                        (short)0, acc, false, false);
        }
        int rbase = krow * 16 + half * 8;
#pragma unroll
        for (int vv = 0; vv < 8; ++vv)
            Ms[(ab * WK + rbase + vv) * MS_STRIDE + pt + ln] = acc[vv];
    }
    __syncthreads();

    // ---- stage 3: output transform Y(2x2) = AT m AT^T ----------------------
#pragma unroll
    for (int it = 0; it < 8; ++it) {
        int item = it * 256 + tid;
        int pl = item & 31;
        int k  = item >> 5;
        int p  = p0 + pl;
        if (p >= WP) continue;
        int n = p / (WT * WT);
        int r = p % (WT * WT);
        int t = r / WT, uu = r % WT;

        float m[4][4];
#pragma unroll
        for (int a = 0; a < 4; ++a)
#pragma unroll
            for (int b = 0; b < 4; ++b)
                m[a][b] = Ms[((a * 4 + b) * WK + k) * MS_STRIDE + pl];

        float s[2][4];
#pragma unroll
        for (int j = 0; j < 4; ++j) {
            s[0][j] = m[0][j] + m[1][j] + m[2][j];
            s[1][j] = m[1][j] - m[2][j] - m[3][j];
        }
        float y[2][2];
#pragma unroll
        for (int a = 0; a < 2; ++a) {
            y[a][0] = s[a][0] + s[a][1] + s[a][2];
            y[a][1] = s[a][1] - s[a][2] - s[a][3];
        }
        float* yp = Y + (((size_t)(n * WK + k) * WOH) + 2 * t) * WOH + 2 * uu;
        yp[0]       = y[0][0];
        yp[1]       = y[0][1];
        yp[WOH]     = y[1][0];
        yp[WOH + 1] = y[1][1];
    }
}

// ---------------------------------------------------------------------------
extern "C" void kernel_launch(void* const* d_in, const int* in_sizes, int n_in,
                              void* d_out, int out_size, void* d_ws, size_t ws_size,
                              hipStream_t stream) {
    const float* x    = (const float*)d_in[0];
    const float* filt = (const float*)d_in[1];
    float*       Y    = (float*)d_out;
    float*       A2   = (float*)d_ws;        // 16*64*64 floats = 256 KB

    (void)hipFuncSetAttribute(reinterpret_cast<const void*>(wg_fused),
                              hipFuncAttributeMaxDynamicSharedMemorySize,
                              SMEM_BYTES);

    wg_filter<<<(WK * WC + 255) / 256, 256, 0, stream>>>(filt, A2);
    wg_fused<<<NBLK, 256, SMEM_BYTES, stream>>>(x, A2, Y);
}